// GraphSageDQN_3083786518763
// MI455X (gfx1250) — compile-verified
//
#include <hip/hip_runtime.h>
#include <hip/hip_bf16.h>

// ---------------------------------------------------------------------------
// GraphSAGE(2-layer, mean aggr) + DQN head for MI455X (gfx1250, wave32, WMMA)
//
// Math trick: mean(x_j) @ Wl.T == mean(x_j @ Wl.T)  -> project first, then
// aggregate in 64 dims. Halves layer-1 scatter traffic; 32MB tables live in
// the 192MB L2 so the random edge gathers are L2-resident.
// GEMMs use V_WMMA_F32_16X16X4_F32 (exact fp32) — one 16x16 tile per wave.
// ---------------------------------------------------------------------------

#define N_NODES        131072
#define N_EDGES        2097152
#define N_GRAPHS       64
#define NODES_PER_G    2048
#define D_IN           128
#define D_EMB          64
#define D_HID          128
#define N_ACT          2048

typedef __attribute__((ext_vector_type(2))) float v2f;
typedef __attribute__((ext_vector_type(8))) float v8f;

// ---------------------------------------------------------------------------
// Zero-fill (graph-capture-safe replacement for memset)
// ---------------------------------------------------------------------------
__global__ void zero_kernel(float* __restrict__ p, long n) {
    long t = (long)blockIdx.x * blockDim.x + threadIdx.x;
    long stride = (long)gridDim.x * blockDim.x;
    for (long i = t; i < n; i += stride) p[i] = 0.0f;
}

// ---------------------------------------------------------------------------
// In-degree count (float, matches reference segment_sum of ones)
// ---------------------------------------------------------------------------
__global__ void degree_kernel(const int* __restrict__ dst,
                              float* __restrict__ cnt, int nEdges) {
    int e = blockIdx.x * blockDim.x + threadIdx.x;
    if (e < nEdges) atomicAdd(&cnt[dst[e]], 1.0f);
}

// ---------------------------------------------------------------------------
// Dual GEMM via fp32 WMMA:  P = In @ Wl^T   and   Q = In @ Wr^T
//   In: [nRows, K] row-major,  Wl/Wr: [64, K] row-major,  P/Q: [nRows, 64]
// grid.x = nRows/16 (M tiles), block = 128 threads = 4 waves, wave w -> n0=16w
// A-frag (16x4 f32): lane l(0..15), half h: A[m0+l][k+2h .. k+2h+1]  (float2)
// B-frag ( 4x16)   : lane l, half h:       W[n0+l][k+2h .. k+2h+1]  (float2)
// C/D 16x16 f32    : VGPR r, lane l, half h -> row m0+r+8h, col n0+l
// EXEC is all-1s: no divergence, dims divide exactly.
// ---------------------------------------------------------------------------
__global__ void sage_dual_gemm_kernel(const float* __restrict__ In,
                                      const float* __restrict__ Wl,
                                      const float* __restrict__ Wr,
                                      float* __restrict__ P,
                                      float* __restrict__ Q,
                                      int K) {
    const int lane = threadIdx.x & 31;
    const int wave = threadIdx.x >> 5;
    const int l    = lane & 15;
    const int h    = lane >> 4;          // lane half: 0 or 1
    const int m0   = blockIdx.x << 4;
    const int n0   = wave << 4;

    const float* arow  = In + (size_t)(m0 + l) * K;
    const float* blrow = Wl + (size_t)(n0 + l) * K;
    const float* brrow = Wr + (size_t)(n0 + l) * K;

    v8f accP = {};
    v8f accQ = {};

    #pragma unroll 4
    for (int k = 0; k < K; k += 4) {
        const int kk = k + 2 * h;        // 8-byte aligned
        v2f a  = *(const v2f*)(arow  + kk);
        v2f bl = *(const v2f*)(blrow + kk);
        v2f br = *(const v2f*)(brrow + kk);
        accP = __builtin_amdgcn_wmma_f32_16x16x4_f32(
            false, a, false, bl, (short)0, accP, false, false);
        accQ = __builtin_amdgcn_wmma_f32_16x16x4_f32(
            false, a, false, br, (short)0, accQ, false, false);
    }

    #pragma unroll
    for (int r = 0; r < 8; ++r) {
        const size_t row = (size_t)(m0 + r + 8 * h);
        P[row * D_EMB + n0 + l] = accP[r];
        Q[row * D_EMB + n0 + l] = accQ[r];
    }
}

// ---------------------------------------------------------------------------
// Edge scatter-add: agg[dst] += V[src]  over 64 dims.
// 16 threads per edge, float4 per thread -> global_atomic_add_f32 x4.
// ---------------------------------------------------------------------------
__global__ void scatter_add_kernel(const float* __restrict__ V,
                                   const int* __restrict__ src,
                                   const int* __restrict__ dst,
                                   float* __restrict__ agg, int nEdges) {
    long t = (long)blockIdx.x * blockDim.x + threadIdx.x;
    long e = t >> 4;
    if (e >= nEdges) return;
    int d  = (int)(t & 15) << 2;
    int s  = src[e];
    int dd = dst[e];
    const float4 v = *(const float4*)(V + (size_t)s * D_EMB + d);
    float* out = agg + (size_t)dd * D_EMB + d;
    atomicAdd(out + 0, v.x);
    atomicAdd(out + 1, v.y);
    atomicAdd(out + 2, v.z);
    atomicAdd(out + 3, v.w);
}

// ---------------------------------------------------------------------------
// Finalize: out = agg / max(cnt,1) + bias + base     (per node x 64 dims)
// ---------------------------------------------------------------------------
__global__ void finalize_kernel(const float* __restrict__ agg,
                                const float* __restrict__ cnt,
                                const float* __restrict__ bias,
                                const float* __restrict__ base,
                                float* __restrict__ out, long nNodes) {
    long t = (long)blockIdx.x * blockDim.x + threadIdx.x;
    long node = t >> 6;
    if (node >= nNodes) return;
    int d = (int)(t & 63);
    float c = fmaxf(cnt[node], 1.0f);
    out[t] = agg[t] / c + bias[d] + base[t];
}

// ---------------------------------------------------------------------------
// Per-graph mean pool: g[b][d] = mean_{node} h2[b*2048+node][d]
// One block per graph; 256 threads = 64 dims x 4 slices + LDS reduce.
// ---------------------------------------------------------------------------
__global__ void pool_kernel(const float* __restrict__ h2,
                            float* __restrict__ g) {
    __shared__ float red[256];
    const int b = blockIdx.x;
    const int d = threadIdx.x & 63;
    const int s = threadIdx.x >> 6;
    const float* base = h2 + (size_t)b * NODES_PER_G * D_EMB;
    float acc = 0.0f;
    for (int node = s; node < NODES_PER_G; node += 4)
        acc += base[(size_t)node * D_EMB + d];
    red[threadIdx.x] = acc;
    __syncthreads();
    if (s == 0) {
        float sum = red[d] + red[64 + d] + red[128 + d] + red[192 + d];
        g[b * D_EMB + d] = sum * (1.0f / (float)NODES_PER_G);
    }
}

// ---------------------------------------------------------------------------
// Head FC1: h = relu(g @ fc1_W^T + fc1_b)   [64,128]  (tiny, scalar is fine)
// ---------------------------------------------------------------------------
__global__ void head1_kernel(const float* __restrict__ g,
                             const float* __restrict__ W,
                             const float* __restrict__ b,
                             float* __restrict__ h) {
    const int i = blockIdx.x;     // 64 rows
    const int j = threadIdx.x;    // 128 cols
    float acc = b[j];
    #pragma unroll 8
    for (int k = 0; k < D_EMB; ++k)
        acc += g[i * D_EMB + k] * W[j * D_EMB + k];
    h[i * D_HID + j] = fmaxf(acc, 0.0f);
}

// ---------------------------------------------------------------------------
// Head FC2: q = h @ fc2_W^T + fc2_b   [64,2048]
// ---------------------------------------------------------------------------
__global__ void head2_kernel(const float* __restrict__ h,
                             const float* __restrict__ W,
                             const float* __restrict__ b,
                             float* __restrict__ q) {
    long t = (long)blockIdx.x * blockDim.x + threadIdx.x;
    if (t >= (long)N_GRAPHS * N_ACT) return;
    const int i = (int)(t >> 11);      // graph
    const int j = (int)(t & 2047);     // action
    float acc = b[j];
    #pragma unroll 8
    for (int k = 0; k < D_HID; ++k)
        acc += h[i * D_HID + k] * W[j * D_HID + k];
    q[t] = acc;
}

// ---------------------------------------------------------------------------
// Launcher
// ---------------------------------------------------------------------------
extern "C" void kernel_launch(void* const* d_in, const int* in_sizes, int n_in,
                              void* d_out, int out_size, void* d_ws, size_t ws_size,
                              hipStream_t stream) {
    const float* x     = (const float*)d_in[0];
    const int*   eidx  = (const int*)d_in[1];
    const float* W1l   = (const float*)d_in[2];
    const float* b1l   = (const float*)d_in[3];
    const float* W1r   = (const float*)d_in[4];
    const float* W2l   = (const float*)d_in[5];
    const float* b2l   = (const float*)d_in[6];
    const float* W2r   = (const float*)d_in[7];
    const float* fc1W  = (const float*)d_in[8];
    const float* fc1b  = (const float*)d_in[9];
    const float* fc2W  = (const float*)d_in[10];
    const float* fc2b  = (const float*)d_in[11];
    const int* src = eidx;             // edge_index[0]
    const int* dst = eidx + N_EDGES;   // edge_index[1]

    // Workspace layout (bytes)
    const size_t SLAB = (size_t)N_NODES * D_EMB * sizeof(float);  // 32 MB
    char* ws = (char*)d_ws;
    float* bufA = (float*)(ws);                 // xl (both layers), then h2
    float* bufB = (float*)(ws + 1 * SLAB);      // base (both layers)
    float* bufC = (float*)(ws + 2 * SLAB);      // agg (zeroed per layer)
    float* bufD = (float*)(ws + 3 * SLAB);      // h1
    float* cnt  = (float*)(ws + 4 * SLAB);      // [N_NODES]
    float* g    = (float*)(ws + 4 * SLAB + (size_t)N_NODES * sizeof(float));
    float* hh   = g + N_GRAPHS * D_EMB;
    float* q    = (float*)d_out;

    const long nodeElems = (long)N_NODES * D_EMB;

    // --- degree (shared by both layers) ---
    zero_kernel<<<512, 256, 0, stream>>>(cnt, N_NODES);
    degree_kernel<<<(N_EDGES + 255) / 256, 256, 0, stream>>>(dst, cnt, N_EDGES);

    // --- layer 1: xl = x@W1l^T, base = x@W1r^T (WMMA fp32) ---
    sage_dual_gemm_kernel<<<N_NODES / 16, 128, 0, stream>>>(
        x, W1l, W1r, bufA, bufB, D_IN);
    zero_kernel<<<2048, 256, 0, stream>>>(bufC, nodeElems);
    scatter_add_kernel<<<(N_EDGES * 16) / 256, 256, 0, stream>>>(
        bufA, src, dst, bufC, N_EDGES);
    finalize_kernel<<<(int)(nodeElems / 256), 256, 0, stream>>>(
        bufC, cnt, b1l, bufB, bufD, N_NODES);          // bufD = h1

    // --- layer 2: xl = h1@W2l^T, base = h1@W2r^T ---
    sage_dual_gemm_kernel<<<N_NODES / 16, 128, 0, stream>>>(
        bufD, W2l, W2r, bufA, bufB, D_EMB);
    zero_kernel<<<2048, 256, 0, stream>>>(bufC, nodeElems);
    scatter_add_kernel<<<(N_EDGES * 16) / 256, 256, 0, stream>>>(
        bufA, src, dst, bufC, N_EDGES);
    finalize_kernel<<<(int)(nodeElems / 256), 256, 0, stream>>>(
        bufC, cnt, b2l, bufB, bufA, N_NODES);          // bufA = h2

    // --- pooling + head ---
    pool_kernel<<<N_GRAPHS, 256, 0, stream>>>(bufA, g);
    head1_kernel<<<N_GRAPHS, D_HID, 0, stream>>>(g, fc1W, fc1b, hh);
    head2_kernel<<<(N_GRAPHS * N_ACT + 255) / 256, 256, 0, stream>>>(
        hh, fc2W, fc2b, q);
}